// Clusterize_24584392802547
// MI455X (gfx1250) — compile-verified
//
#include <hip/hip_runtime.h>
#include <stdint.h>

// Connected-components labeling (4-connectivity) for 256 images of 256x256.
// Final label per component = min linear pixel index in component + 1 (bg=0),
// the unique fixpoint of the reference's iteration.
//
// CDNA5-specific usage:
//  - whole 256KB int32 label image resident in LDS (320KB/WGP on gfx1250)
//  - global_load_async_to_lds_b128  (ASYNCcnt DMA, global -> LDS) for input
//  - global_store_async_from_lds_b128 (ASYNCcnt DMA, LDS -> global) for output
//  - s_wait_asynccnt fencing

#define SDIM     256
#define PIX      (SDIM * SDIM)   // 65536 pixels
#define THREADS  1024
#define PPT      (PIX / THREADS) // 64 pixels per thread
#define CHUNKS   ((PIX * 4) / (THREADS * 16))  // 16 x b128 issues per thread

__global__ __launch_bounds__(THREADS, 1)
void cc_label_kernel(const float* __restrict__ in, float* __restrict__ out)
{
    extern __shared__ int lds[];        // PIX labels (256KB) + change flag
    int* lab  = lds;
    int* flag = lds + PIX;

    const int tid = threadIdx.x;
    const int img = blockIdx.x;
    const float* src = in + (size_t)img * PIX;
    float*       dst = out + (size_t)img * PIX;

    // low 32 bits of a flat shared pointer == LDS byte offset
    const unsigned ldsBase = (unsigned)(uintptr_t)(void*)lab;

    // ---------------- async DMA: global -> LDS (256KB) ----------------
    // Each lane moves 16B per issue; 16 issues per thread cover the image.
    {
        #pragma unroll
        for (int k = 0; k < CHUNKS; ++k) {
            unsigned byteOff = (unsigned)(tid * 16 + k * THREADS * 16);
            unsigned ldsAddr = ldsBase + byteOff;
            asm volatile("global_load_async_to_lds_b128 %0, %1, %2"
                         :: "v"(ldsAddr), "v"(byteOff), "s"(src)
                         : "memory");
        }
        asm volatile("s_wait_asynccnt 0" ::: "memory");
    }
    __syncthreads();   // every wave waited its own DMA before anyone reads

    // ---------------- init labels in place: fg -> p+1, bg -> 0 --------
    #pragma unroll 4
    for (int k = 0; k < PPT; ++k) {
        int p = tid + k * THREADS;
        float v = __uint_as_float((unsigned)lab[p]);
        lab[p] = (v != 0.0f) ? (p + 1) : 0;
    }
    __syncthreads();

    // ---------------- iterate: neighbor-min + path compression --------
    // Labels only decrease; invariant: lab[p]-1 is a pixel index inside p's
    // component => unique fixpoint = (min index in component) + 1, which is
    // exactly the reference's converged answer.
    for (;;) {
        if (tid == 0) *flag = 0;
        __syncthreads();

        int any = 0;
        #pragma unroll 2
        for (int k = 0; k < PPT; ++k) {
            int p = tid + k * THREADS;
            int l = lab[p];
            if (l == 0) continue;
            int x = p & (SDIM - 1);
            int y = p >> 8;
            int m = l;
            if (y > 0)        { int v = lab[p - SDIM]; if (v && v < m) m = v; }
            if (y < SDIM - 1) { int v = lab[p + SDIM]; if (v && v < m) m = v; }
            if (x > 0)        { int v = lab[p - 1];    if (v && v < m) m = v; }
            if (x < SDIM - 1) { int v = lab[p + 1];    if (v && v < m) m = v; }
            // chase to current root (full path compression; labels strictly
            // decrease along the chain, so this terminates)
            int h = lab[m - 1];
            while (h != 0 && h < m) { m = h; h = lab[m - 1]; }
            if (m < l) { lab[p] = m; any = 1; }
        }
        if (any) *flag = 1;      // benign same-value race
        __syncthreads();

        int done = (*flag == 0);
        __syncthreads();         // protect flag read vs next reset
        if (done) break;
    }

    // ------- convert labels to float32 in place (owner-only, no race) -
    #pragma unroll 4
    for (int k = 0; k < PPT; ++k) {
        int p = tid + k * THREADS;
        lab[p] = (int)__float_as_uint((float)lab[p]);
    }
    __syncthreads();   // conversion visible before the DMA engine reads LDS

    // ---------------- async DMA: LDS -> global (256KB) ----------------
    {
        #pragma unroll
        for (int k = 0; k < CHUNKS; ++k) {
            unsigned byteOff = (unsigned)(tid * 16 + k * THREADS * 16);
            unsigned ldsAddr = ldsBase + byteOff;
            asm volatile("global_store_async_from_lds_b128 %0, %1, %2"
                         :: "v"(byteOff), "v"(ldsAddr), "s"(dst)
                         : "memory");
        }
        asm volatile("s_wait_asynccnt 0" ::: "memory");
    }
}

extern "C" void kernel_launch(void* const* d_in, const int* in_sizes, int n_in,
                              void* d_out, int out_size, void* d_ws, size_t ws_size,
                              hipStream_t stream)
{
    (void)n_in; (void)d_ws; (void)ws_size; (void)out_size;
    const float* in = (const float*)d_in[0];
    float* out = (float*)d_out;
    int nImg = in_sizes[0] / PIX;     // B*N = 256 images
    size_t shmem = (size_t)PIX * sizeof(int) + 16;   // labels + flag
    cc_label_kernel<<<nImg, THREADS, shmem, stream>>>(in, out);
}